// MultiScaleRecall_34548716929311
// MI455X (gfx1250) — compile-verified
//
#include <hip/hip_runtime.h>

// MultiScaleRecall: per-row masked 1D convs (K=3,5,7) -> softmax -> weighted sum -> mean.
// B = S = 4096, fp32. 64 MiB traffic -> ~2.9 us HBM floor on MI455X; VALU is the real
// limiter, so conv math is packed into v_pk_fma_f32 pairs where the backend allows.
// CDNA5 paths: GLOBAL_LOAD_ASYNC_TO_LDS_B128 (GVS form, immediate-offset chunks) +
// s_wait_asynccnt, per-wave in-order LDS pipe (no barriers), wave32 shfl reduction.

#define S_LEN 4096
#define PAD   4                      // >= max conv halo (3), keeps b128 alignment
#define ROWF  (PAD + S_LEN + PAD)    // 4104 floats per row slice
#define WAVES_PER_BLOCK 2
#define BLOCK (WAVES_PER_BLOCK * 32)

typedef float v2f __attribute__((ext_vector_type(2)));

// 32 chunks x 512 B: LDS dest and global src both advance by the inst offset
// (ISA 15.18.3: the offset is added to both the LDS and the global address),
//  so one VGPR pair serves all 32 loads.
template <int I>
__device__ __forceinline__ void async_row_copy(unsigned lds, unsigned voff,
                                               unsigned long long base) {
  asm volatile("global_load_async_to_lds_b128 %0, %1, %2 offset:%3"
               :: "v"(lds), "v"(voff), "s"(base), "i"(I * 512)
               : "memory");
  if constexpr (I + 1 < 32) async_row_copy<I + 1>(lds, voff, base);
}

__global__ __launch_bounds__(BLOCK) void msr_kernel(
    const float* __restrict__ x,
    const int*   __restrict__ length,
    const float* __restrict__ w1,
    const float* __restrict__ w2,
    const float* __restrict__ w3,
    float*       __restrict__ out,
    int nrows)
{
  __shared__ __align__(16) float rows[WAVES_PER_BLOCK][ROWF];

  const int tid  = threadIdx.x;
  const int wv   = tid >> 5;
  const int lane = tid & 31;
  const int r    = blockIdx.x * WAVES_PER_BLOCK + wv;
  if (r >= nrows) return;                    // grid sized exactly; no barriers in kernel

  float* rowp = rows[wv];
  const int len = length[r];

  // ---- 1) async-copy the 16 KB row into this wave's LDS slice: 32 x b128 per lane ----
  {
    const unsigned lds  = (unsigned)(size_t)rowp + PAD * 4u + (unsigned)lane * 16u;
    const unsigned voff = (unsigned)lane * 16u;
    const unsigned long long base = (unsigned long long)(size_t)(x + (size_t)r * S_LEN);
    async_row_copy<0>(lds, voff, base);
  }
  asm volatile("s_wait_asynccnt 0x0" ::: "memory");

  // ---- 2) zero guards + masked tail (per-wave LDS pipe is in-order; no barrier) ----
  if (lane < PAD)                   rowp[lane] = 0.0f;
  if (lane >= PAD && lane < 2*PAD)  rowp[PAD + S_LEN + (lane - PAD)] = 0.0f;
  for (int p = len + lane; p < S_LEN; p += 32) rowp[PAD + p] = 0.0f;

  // ---- 3) weights, prescaled by log2(e) so exp(l) == exp2(l'): one v_exp_f32 each ----
  const float L2E = 1.4426950408889634f;
  float W1[3], W2[5], W3[7];
  #pragma unroll
  for (int i = 0; i < 3; ++i) W1[i] = w1[i] * L2E;
  #pragma unroll
  for (int i = 0; i < 5; ++i) W2[i] = w2[i] * L2E;
  #pragma unroll
  for (int i = 0; i < 7; ++i) W3[i] = w3[i] * L2E;

  // ---- 4) main loop: 4 contiguous elements/lane/iter, packed-pair f32 math ----
  v2f D1 = {0.f, 0.f}, N1 = {0.f, 0.f};
  v2f D2 = {0.f, 0.f}, N2 = {0.f, 0.f};
  v2f D3 = {0.f, 0.f}, N3 = {0.f, 0.f};
  const float4* rp4 = (const float4*)rowp;

  #pragma unroll 2
  for (int j = 0; j < 32; ++j) {
    const int p0 = (j * 32 + lane) * 4;      // element index, multiple of 4
    const int b  = p0 >> 2;                  // (PAD + p0 - 4)/4 == p0/4
    float4 va = rp4[b];                      // x[p0-4 .. p0-1]
    float4 vb = rp4[b + 1];                  // x[p0   .. p0+3]
    float4 vc = rp4[b + 2];                  // x[p0+4 .. p0+7]
    const float win[12] = { va.x, va.y, va.z, va.w,
                            vb.x, vb.y, vb.z, vb.w,
                            vc.x, vc.y, vc.z, vc.w };  // win[i] = x[p0-4+i]
    // adjacent-element pairs: Pi = { win[i], win[i+1] }
    const v2f P1 = {win[1], win[2]}, P2 = {win[2], win[3]}, P3 = {win[3], win[4]};
    const v2f P4 = {win[4], win[5]}, P5 = {win[5], win[6]}, P6 = {win[6], win[7]};
    const v2f P7 = {win[7], win[8]}, P8 = {win[8], win[9]}, P9 = {win[9], win[10]};

    // logits for output pairs (k=0,1) -> *a and (k=2,3) -> *b ; contraction -> pk_fma
    v2f l1a = W1[0]*P3 + W1[1]*P4 + W1[2]*P5;
    v2f l1b = W1[0]*P5 + W1[1]*P6 + W1[2]*P7;
    v2f l2a = W2[0]*P2 + W2[1]*P3 + W2[2]*P4 + W2[3]*P5 + W2[4]*P6;
    v2f l2b = W2[0]*P4 + W2[1]*P5 + W2[2]*P6 + W2[3]*P7 + W2[4]*P8;
    v2f l3a = W3[0]*P1 + W3[1]*P2 + W3[2]*P3 + W3[3]*P4 + W3[4]*P5 + W3[5]*P6 + W3[6]*P7;
    v2f l3b = W3[0]*P3 + W3[1]*P4 + W3[2]*P5 + W3[3]*P6 + W3[4]*P7 + W3[5]*P8 + W3[6]*P9;

    const v2f e1a = { __builtin_amdgcn_exp2f(l1a.x), __builtin_amdgcn_exp2f(l1a.y) };
    const v2f e1b = { __builtin_amdgcn_exp2f(l1b.x), __builtin_amdgcn_exp2f(l1b.y) };
    const v2f e2a = { __builtin_amdgcn_exp2f(l2a.x), __builtin_amdgcn_exp2f(l2a.y) };
    const v2f e2b = { __builtin_amdgcn_exp2f(l2b.x), __builtin_amdgcn_exp2f(l2b.y) };
    const v2f e3a = { __builtin_amdgcn_exp2f(l3a.x), __builtin_amdgcn_exp2f(l3a.y) };
    const v2f e3b = { __builtin_amdgcn_exp2f(l3b.x), __builtin_amdgcn_exp2f(l3b.y) };

    D1 += e1a + e1b;  N1 = N1 + e1a * P4 + e1b * P6;   // xv pairs are P4 / P6
    D2 += e2a + e2b;  N2 = N2 + e2a * P4 + e2b * P6;
    D3 += e3a + e3b;  N3 = N3 + e3a * P4 + e3b * P6;
  }

  float d1 = D1.x + D1.y, n1 = N1.x + N1.y;
  float d2 = D2.x + D2.y, n2 = N2.x + N2.y;
  float d3 = D3.x + D3.y, n3 = N3.x + N3.y;

  // ---- 5) denominator fix-up: tail (p >= len) contributed exp(l_p); past the halo
  //      the zeroed tail gives exp(0)=1 exactly, the <=3 boundary slots get exp(l)-1.
  //      Numerators needed no fix (x == 0 on the whole tail). ----
  if (lane < 3) {
    const int p = len + lane;
    if (p < S_LEN) {
      float z[7];
      #pragma unroll
      for (int d = 0; d < 7; ++d) z[d] = rowp[PAD + p - 3 + d];   // z[d] = x[p-3+d]
      float l1 = W1[0]*z[2]; l1 = fmaf(W1[1], z[3], l1); l1 = fmaf(W1[2], z[4], l1);
      float l2 = W2[0]*z[1];
      #pragma unroll
      for (int q = 1; q < 5; ++q) l2 = fmaf(W2[q], z[1 + q], l2);
      float l3 = W3[0]*z[0];
      #pragma unroll
      for (int q = 1; q < 7; ++q) l3 = fmaf(W3[q], z[q], l3);
      d1 -= (__builtin_amdgcn_exp2f(l1) - 1.0f);
      d2 -= (__builtin_amdgcn_exp2f(l2) - 1.0f);
      d3 -= (__builtin_amdgcn_exp2f(l3) - 1.0f);
    }
  }

  // ---- 6) wave32 butterfly reduction of 6 accumulators ----
  #pragma unroll
  for (int off = 16; off >= 1; off >>= 1) {
    d1 += __shfl_xor(d1, off, 32);
    n1 += __shfl_xor(n1, off, 32);
    d2 += __shfl_xor(d2, off, 32);
    n2 += __shfl_xor(n2, off, 32);
    d3 += __shfl_xor(d3, off, 32);
    n3 += __shfl_xor(n3, off, 32);
  }

  if (lane == 0) {
    const float tail = (float)(S_LEN - len);   // exp(0)=1 contributions past the halo
    d1 -= tail; d2 -= tail; d3 -= tail;
    out[r] = (n1 / d1 + n2 / d2 + n3 / d3) * (1.0f / 3.0f);
  }
}

extern "C" void kernel_launch(void* const* d_in, const int* in_sizes, int n_in,
                              void* d_out, int out_size, void* d_ws, size_t ws_size,
                              hipStream_t stream) {
  const float* x   = (const float*)d_in[0];
  const int*   len = (const int*)  d_in[1];
  const float* w1  = (const float*)d_in[2];
  const float* w2  = (const float*)d_in[3];
  const float* w3  = (const float*)d_in[4];
  float*       out = (float*)d_out;
  const int B = in_sizes[1];                       // length has B entries
  const int grid = (B + WAVES_PER_BLOCK - 1) / WAVES_PER_BLOCK;
  msr_kernel<<<grid, BLOCK, 0, stream>>>(x, len, w1, w2, w3, out, B);
}